// CustomYOLOLoss_71201967833741
// MI455X (gfx1250) — compile-verified
//
#include <hip/hip_runtime.h>
#include <cstdint>

// ---------------------------------------------------------------------------
// YOLO loss for MI455X (gfx1250): pure bandwidth-bound streaming reduction.
// ~111 MB in -> 4 scalars out.  Uses CDNA5 async global->LDS DMA (ASYNCcnt)
// with double-buffered 256-cell tiles, wave32 shuffle reductions, and a
// deterministic two-kernel reduction (no float atomics).
// ---------------------------------------------------------------------------

#define TILE     256
#define NTHREADS 256
#define GRID_MAX 1024
#define EPSF     1e-7f

typedef int v4i_t __attribute__((ext_vector_type(4)));

// ---- CDNA5 async copy: global -> LDS, 16 bytes, tracked by ASYNCcnt -------
__device__ __forceinline__ void async_copy16(const void* g, void* l) {
#if __has_builtin(__builtin_amdgcn_global_load_async_to_lds_b128)
  typedef __attribute__((address_space(1))) v4i_t* gp_t;
  typedef __attribute__((address_space(3))) v4i_t* lp_t;
  __builtin_amdgcn_global_load_async_to_lds_b128(
      (gp_t)(uintptr_t)g, (lp_t)(uintptr_t)l, /*offset=*/0, /*cpol=*/0);
#else
  unsigned lds_off = (unsigned)(uintptr_t)l;  // AS3 offset = low 32 bits
  asm volatile("global_load_async_to_lds_b128 %0, %1, off"
               :: "v"(lds_off), "v"(g) : "memory");
#endif
}

template <int N>
__device__ __forceinline__ void wait_async() {
#if __has_builtin(__builtin_amdgcn_s_wait_asynccnt)
  __builtin_amdgcn_s_wait_asynccnt(N);
#else
  asm volatile("s_wait_asynccnt %0" :: "i"(N) : "memory");
#endif
}

// ---- math helpers ---------------------------------------------------------
__device__ __forceinline__ float fast_rcp(float x) {
  return __builtin_amdgcn_rcpf(x);
}
__device__ __forceinline__ float sigmoidf(float x) {
  return fast_rcp(1.0f + __expf(-x));
}
__device__ __forceinline__ float bce0f(float x) {  // bce_logits(x, 0)
  return fmaxf(x, 0.0f) + log1pf(__expf(-fabsf(x)));
}

// Per-cell evaluation: 3 boxes, 5 channels each; target 5 channels.
__device__ __forceinline__ void accum_cell(const float* __restrict__ ci,
                                           const float* __restrict__ ct,
                                           float& s_noobj, float& s_nresp,
                                           float& s_bbox, float& s_obj,
                                           float& s_nobj) {
  float obj = (ct[0] > 0.0f) ? 1.0f : 0.0f;
  float thw = ct[3] * 0.5f, thh = ct[4] * 0.5f;
  float tx1 = ct[1] - thw, ty1 = ct[2] - thh;
  float tx2 = ct[1] + thw, ty2 = ct[2] + thh;
  float tarea = fmaxf(tx2 - tx1, 0.0f) * fmaxf(ty2 - ty1, 0.0f);

  float bce0_sum = 0.0f;
  float best_iou = -1.0f;
  float b_x1 = 0.f, b_y1 = 0.f, b_x2 = 0.f, b_y2 = 0.f;
  float b_union = 0.f, b_bce0 = 0.f, b_l0 = 0.f;

#pragma unroll
  for (int b = 0; b < 3; ++b) {
    float l0 = ci[5 * b + 0];
    float bce0 = bce0f(l0);
    bce0_sum += bce0;
    float px = sigmoidf(ci[5 * b + 1]);
    float py = sigmoidf(ci[5 * b + 2]);
    float pw = sigmoidf(ci[5 * b + 3]) * 0.5f;
    float ph = sigmoidf(ci[5 * b + 4]) * 0.5f;
    float x1 = px - pw, y1 = py - ph, x2 = px + pw, y2 = py + ph;
    float lx = fmaxf(x1, tx1), ly = fmaxf(y1, ty1);
    float rx = fminf(x2, tx2), ry = fminf(y2, ty2);
    float inter = fmaxf(rx - lx, 0.0f) * fmaxf(ry - ly, 0.0f);
    float parea = fmaxf(x2 - x1, 0.0f) * fmaxf(y2 - y1, 0.0f);
    float uni = parea + tarea - inter;
    float iou = inter * fast_rcp(uni + EPSF);
    bool better = iou > best_iou;  // strict: first max wins (jnp.argmax)
    best_iou = better ? iou : best_iou;
    b_x1 = better ? x1 : b_x1;  b_y1 = better ? y1 : b_y1;
    b_x2 = better ? x2 : b_x2;  b_y2 = better ? y2 : b_y2;
    b_union = better ? uni : b_union;
    b_bce0 = better ? bce0 : b_bce0;
    b_l0 = better ? l0 : b_l0;
  }
  float ex1 = fminf(b_x1, tx1), ey1 = fminf(b_y1, ty1);
  float ex2 = fmaxf(b_x2, tx2), ey2 = fmaxf(b_y2, ty2);
  float enc = fmaxf(ex2 - ex1, 0.0f) * fmaxf(ey2 - ey1, 0.0f);
  float giou = best_iou - (enc - b_union) * fast_rcp(enc + EPSF);

  s_noobj += (1.0f - obj) * bce0_sum;
  s_nresp += obj * (bce0_sum - b_bce0);
  s_bbox  += obj * (1.0f - giou);
  s_obj   += obj * (b_bce0 - b_l0);  // bce_logits(x,1) = bce0(x) - x
  s_nobj  += obj;
}

// Issue exactly 5 async b128 loads per thread: 960 f4 input + 320 f4 target.
// Branch boundaries fall on wave boundaries, so each wave issues exactly 5
// async instructions per tile (uniform ASYNCcnt).
__device__ __forceinline__ void stage_tile(const float* __restrict__ gin,
                                           const float* __restrict__ gtg,
                                           long long tile, float* lin,
                                           float* ltg, int tid) {
  const char* gi = (const char*)(gin + tile * (long long)(TILE * 15));
  const char* gt = (const char*)(gtg + tile * (long long)(TILE * 5));
#pragma unroll
  for (int j = 0; j < 5; ++j) {
    int idx = j * NTHREADS + tid;
    if (idx < 960) {
      async_copy16(gi + (size_t)idx * 16, (char*)lin + (size_t)idx * 16);
    } else {
      int k = idx - 960;
      async_copy16(gt + (size_t)k * 16, (char*)ltg + (size_t)k * 16);
    }
  }
}

__global__ __launch_bounds__(NTHREADS) void yolo_partials(
    const float* __restrict__ gin, const float* __restrict__ gtg,
    float* __restrict__ ws, long long cells, long long tiles) {
  __shared__ float lin[2][TILE * 15];
  __shared__ float ltg[2][TILE * 5];
  __shared__ float red[NTHREADS / 32][5];

  const int tid = threadIdx.x;
  float s1 = 0.f, s2 = 0.f, s3 = 0.f, s4 = 0.f, s5 = 0.f;

  long long t0 = blockIdx.x;
  int cur = 0;
  if (t0 < tiles) stage_tile(gin, gtg, t0, lin[0], ltg[0], tid);

  for (long long t = t0; t < tiles; t += gridDim.x) {
    long long nxt = t + gridDim.x;
    bool hasNext = nxt < tiles;
    if (hasNext) stage_tile(gin, gtg, nxt, lin[cur ^ 1], ltg[cur ^ 1], tid);
    // Async loads complete in order: <=5 outstanding => current tile resident.
    if (hasNext) wait_async<5>(); else wait_async<0>();
    __syncthreads();
    accum_cell(&lin[cur][tid * 15], &ltg[cur][tid * 5], s1, s2, s3, s4, s5);
    __syncthreads();  // all readers done before this buffer is restaged
    cur ^= 1;
  }

  // Tail cells (cells % TILE) via direct loads (empty for the ref shape).
  for (long long c = tiles * TILE + (long long)blockIdx.x * NTHREADS + tid;
       c < cells; c += (long long)gridDim.x * NTHREADS) {
    float ci[15], ct[5];
    const float* gi = gin + c * 15;
    const float* gt = gtg + c * 5;
#pragma unroll
    for (int k = 0; k < 15; ++k) ci[k] = gi[k];
#pragma unroll
    for (int k = 0; k < 5; ++k) ct[k] = gt[k];
    accum_cell(ci, ct, s1, s2, s3, s4, s5);
  }

  // wave32 butterfly reduction
#pragma unroll
  for (int off = 16; off > 0; off >>= 1) {
    s1 += __shfl_xor(s1, off, 32);
    s2 += __shfl_xor(s2, off, 32);
    s3 += __shfl_xor(s3, off, 32);
    s4 += __shfl_xor(s4, off, 32);
    s5 += __shfl_xor(s5, off, 32);
  }
  const int wave = tid >> 5, lane = tid & 31;
  __syncthreads();
  if (lane == 0) {
    red[wave][0] = s1; red[wave][1] = s2; red[wave][2] = s3;
    red[wave][3] = s4; red[wave][4] = s5;
  }
  __syncthreads();
  if (tid == 0) {
    float r0 = 0, r1 = 0, r2 = 0, r3 = 0, r4 = 0;
#pragma unroll
    for (int w = 0; w < NTHREADS / 32; ++w) {
      r0 += red[w][0]; r1 += red[w][1]; r2 += red[w][2];
      r3 += red[w][3]; r4 += red[w][4];
    }
    float* o = ws + (size_t)blockIdx.x * 5;
    o[0] = r0; o[1] = r1; o[2] = r2; o[3] = r3; o[4] = r4;
  }
}

__global__ __launch_bounds__(NTHREADS) void yolo_finalize(
    const float* __restrict__ ws, int nblocks, float total_cells,
    float* __restrict__ out) {
  __shared__ float red[NTHREADS / 32][5];
  const int tid = threadIdx.x;
  float a0 = 0, a1 = 0, a2 = 0, a3 = 0, a4 = 0;
  for (int i = tid; i < nblocks; i += NTHREADS) {
    const float* p = ws + (size_t)i * 5;
    a0 += p[0]; a1 += p[1]; a2 += p[2]; a3 += p[3]; a4 += p[4];
  }
#pragma unroll
  for (int off = 16; off > 0; off >>= 1) {
    a0 += __shfl_xor(a0, off, 32);
    a1 += __shfl_xor(a1, off, 32);
    a2 += __shfl_xor(a2, off, 32);
    a3 += __shfl_xor(a3, off, 32);
    a4 += __shfl_xor(a4, off, 32);
  }
  const int wave = tid >> 5, lane = tid & 31;
  if (lane == 0) {
    red[wave][0] = a0; red[wave][1] = a1; red[wave][2] = a2;
    red[wave][3] = a3; red[wave][4] = a4;
  }
  __syncthreads();
  if (tid == 0) {
    float r0 = 0, r1 = 0, r2 = 0, r3 = 0, r4 = 0;
#pragma unroll
    for (int w = 0; w < NTHREADS / 32; ++w) {
      r0 += red[w][0]; r1 += red[w][1]; r2 += red[w][2];
      r3 += red[w][3]; r4 += red[w][4];
    }
    float n_obj = r4;
    float n_noobj = total_cells - n_obj;
    float inv_obj = 1.0f / n_obj;
    float loss_noobj = r0 / (n_noobj * 3.0f) + r1 / (n_obj * 2.0f);
    float loss_bbox = r2 * inv_obj;
    float loss_obj = r3 * inv_obj;
    out[0] = loss_obj + loss_bbox + loss_noobj;
    out[1] = loss_noobj;
    out[2] = loss_bbox;
    out[3] = loss_obj;
  }
}

extern "C" void kernel_launch(void* const* d_in, const int* in_sizes, int n_in,
                              void* d_out, int out_size, void* d_ws,
                              size_t ws_size, hipStream_t stream) {
  const float* gin = (const float*)d_in[0];   // (N,S,S,15) float32
  const float* gtg = (const float*)d_in[1];   // (N,S,S,5)  float32
  float* out = (float*)d_out;                 // 4 scalars
  float* ws = (float*)d_ws;

  long long cells = (long long)in_sizes[0] / 15;  // N*S*S
  long long tiles = cells / TILE;
  long long wantG = tiles > 0 ? tiles : 1;
  int G = (int)(wantG < GRID_MAX ? wantG : GRID_MAX);

  yolo_partials<<<G, NTHREADS, 0, stream>>>(gin, gtg, ws, cells, tiles);
  yolo_finalize<<<1, NTHREADS, 0, stream>>>(ws, G, (float)cells, out);
}